// SimpleDiagonalRNN_71047349010753
// MI455X (gfx1250) — compile-verified
//
#include <hip/hip_runtime.h>
#include <stdint.h>

// Diagonal linear RNN scan: h_t = w * h_{t-1} + x_t, h_0 = 0, over [B, L, D].
// Memory-bound (AI ~ 0.25 FLOP/B) -> chunked 3-phase scan to expose enough
// parallelism to saturate 23.3 TB/s HBM; x is streamed through LDS with
// double-buffered CDNA5 async global->LDS copies (ASYNCcnt).
//
// L2 (192 MB) tactics: K1 streams x ascending and leaves the tail resident;
// K3 walks blocks in REVERSE x order so its second read of x hits the hot
// tail, and emits h with non-temporal stores so the 268 MB output stream
// does not evict the x data being reused.

namespace rnn {
constexpr int kB  = 32;
constexpr int kL  = 4096;
constexpr int kD  = 512;                 // one thread per channel d
constexpr int kTC = 256;                 // timesteps per chunk
constexpr int kNC = kL / kTC;            // 16 chunks per sequence
constexpr int kTT = 8;                   // timesteps per LDS tile
constexpr int kNT = kTC / kTT;           // 32 tiles per chunk
constexpr int kTileBytes = kTT * kD * 4; // 16 KiB per tile (x2 buffers = 32 KiB LDS)
constexpr int kQPT = kTT / 4;            // b128 quads per thread per tile (= 2)
static_assert(kTC == 256, "carry scan uses 8 squarings for w^256");
static_assert(kTT * kD * 4 == kQPT * kD * 16, "quad tiling covers tile exactly");
}  // namespace rnn

#define S_WAIT_ASYNCCNT(n) asm volatile("s_wait_asynccnt %0" ::"n"(n) : "memory")

// Issue this thread's share of one tile copy: global (SGPR base + VGPR offset)
// -> LDS, 16 bytes per lane per issue, fully coalesced (512B per wave-issue).
__device__ __forceinline__ void async_copy_tile(const float* __restrict__ gbase,
                                                uint32_t lds_base, int tile,
                                                int buf, int tid) {
#pragma unroll
  for (int k = 0; k < rnn::kQPT; ++k) {
    const uint32_t q    = (uint32_t)(tid + k * rnn::kD);
    const uint32_t goff = (uint32_t)tile * (uint32_t)rnn::kTileBytes + q * 16u;
    const uint32_t loff = lds_base + (uint32_t)buf * (uint32_t)rnn::kTileBytes + q * 16u;
    asm volatile("global_load_async_to_lds_b128 %0, %1, %2"
                 :
                 : "v"(loff), "v"(goff), "s"(gbase)
                 : "memory");
  }
}

// ---------------- Kernel 1: per-chunk partial scan (zero carry-in) ----------
__global__ __launch_bounds__(rnn::kD) void diag_rnn_partials(
    const float* __restrict__ x, const float* __restrict__ w,
    float* __restrict__ carry) {
  using namespace rnn;
  __shared__ float smem[2 * kTT * kD];
  const int tid = threadIdx.x;           // channel d
  const int blk = blockIdx.x;            // (b, chunk), ascending x order
  const int b = blk / kNC;
  const int c = blk % kNC;
  const float* xbase = x + ((size_t)b * kL + (size_t)c * kTC) * kD;  // uniform
  const uint32_t lds0 = (uint32_t)(size_t)(&smem[0]);
  const float wd = w[tid];

  async_copy_tile(xbase, lds0, 0, 0, tid);

  float h = 0.0f;
  for (int tile = 0; tile < kNT; ++tile) {
    if (tile + 1 < kNT) {
      async_copy_tile(xbase, lds0, tile + 1, (tile + 1) & 1, tid);
      S_WAIT_ASYNCCNT(rnn::kQPT);        // drain current tile (in-order), keep next in flight
    } else {
      S_WAIT_ASYNCCNT(0);
    }
    __syncthreads();                     // tile visible to whole workgroup
    const float* buf = &smem[(tile & 1) * kTT * kD];
#pragma unroll
    for (int tt = 0; tt < kTT; ++tt) {
      h = fmaf(wd, h, buf[tt * kD + tid]);
    }
    __syncthreads();                     // done reading before buffer reuse
  }
  carry[((size_t)b * kNC + c) * kD + tid] = h;  // chunk-end state, zero seed
}

// ---------------- Kernel 2: exclusive scan over chunk carries ---------------
__global__ __launch_bounds__(rnn::kD) void diag_rnn_carry_scan(
    const float* __restrict__ w, float* __restrict__ carry) {
  using namespace rnn;
  const int d = threadIdx.x;
  const int b = blockIdx.x;
  float a = w[d];
#pragma unroll
  for (int i = 0; i < 8; ++i) a *= a;    // a = w^256 = w^kTC
  float* base = carry + (size_t)b * kNC * kD + d;
  float h = 0.0f;
#pragma unroll
  for (int c = 0; c < kNC; ++c) {
    const float t = base[(size_t)c * kD];
    base[(size_t)c * kD] = h;            // carry-in for chunk c (exclusive)
    h = fmaf(a, h, t);                   // compose: h_end = w^TC * h_in + partial
  }
}

// ---------------- Kernel 3: rescan chunk with true carry-in, emit states ----
__global__ __launch_bounds__(rnn::kD) void diag_rnn_emit(
    const float* __restrict__ x, const float* __restrict__ w,
    const float* __restrict__ carry, float* __restrict__ out) {
  using namespace rnn;
  __shared__ float smem[2 * kTT * kD];
  const int tid = threadIdx.x;
  // Reverse (b, chunk) order: consume x from the L2-hot tail downward.
  const int idx = (kB * kNC - 1) - (int)blockIdx.x;
  const int b = idx / kNC;
  const int c = idx % kNC;
  const float* xbase = x + ((size_t)b * kL + (size_t)c * kTC) * kD;
  const uint32_t lds0 = (uint32_t)(size_t)(&smem[0]);
  const float wd = w[tid];

  async_copy_tile(xbase, lds0, 0, 0, tid);

  float h = carry[((size_t)b * kNC + c) * kD + tid];  // true carry-in
  for (int tile = 0; tile < kNT; ++tile) {
    if (tile + 1 < kNT) {
      async_copy_tile(xbase, lds0, tile + 1, (tile + 1) & 1, tid);
      S_WAIT_ASYNCCNT(rnn::kQPT);
    } else {
      S_WAIT_ASYNCCNT(0);
    }
    __syncthreads();
    const float* buf = &smem[(tile & 1) * kTT * kD];
    float* orow =
        out + ((size_t)b * kL + (size_t)c * kTC + (size_t)tile * kTT) * kD + tid;
#pragma unroll
    for (int tt = 0; tt < kTT; ++tt) {
      h = fmaf(wd, h, buf[tt * kD + tid]);
      // Never re-read: NT store keeps the 268 MB output stream out of L2.
      __builtin_nontemporal_store(h, &orow[(size_t)tt * kD]);
    }
    __syncthreads();
  }
}

extern "C" void kernel_launch(void* const* d_in, const int* in_sizes, int n_in,
                              void* d_out, int out_size, void* d_ws,
                              size_t ws_size, hipStream_t stream) {
  (void)in_sizes; (void)n_in; (void)out_size; (void)ws_size;
  const float* x = (const float*)d_in[0];   // [B, L, D] fp32
  const float* w = (const float*)d_in[1];   // [D] fp32
  float* out   = (float*)d_out;             // [B, L, D] fp32
  float* carry = (float*)d_ws;              // B*NC*D floats = 1 MiB scratch

  const dim3 blk(rnn::kD);
  diag_rnn_partials<<<dim3(rnn::kB * rnn::kNC), blk, 0, stream>>>(x, w, carry);
  diag_rnn_carry_scan<<<dim3(rnn::kB), blk, 0, stream>>>(w, carry);
  diag_rnn_emit<<<dim3(rnn::kB * rnn::kNC), blk, 0, stream>>>(x, w, carry, out);
}